// ClusterKVAttention_22651657519403
// MI455X (gfx1250) — compile-verified
//
#include <hip/hip_runtime.h>
#include <math.h>

// ---------------- constants from the reference ----------------
#define HID   4096
#define NH    32          // H == HKV
#define DH    128
#define SLEN  32768
#define NLIST 1024
#define SINK  64
#define WIN   64
#define NSEL  2048
#define SMID  (SLEN - SINK - WIN)          // 32640
#define LTOT  (SINK + NSEL + WIN + 1)      // 2177
#define NSPLIT 8

typedef __attribute__((ext_vector_type(2))) float v2f;
typedef __attribute__((ext_vector_type(8))) float v8f;

// ---------------- GEMV: y[row] = dot(W[row,:4096], x) ----------------
// 8 waves / block, one row per wave, float4 coalesced streaming loads.
__global__ void gemv4096(const float* __restrict__ W, const float* __restrict__ x,
                         float* __restrict__ y, int nrows) {
  int row  = blockIdx.x * 8 + (threadIdx.x >> 5);
  int lane = threadIdx.x & 31;
  if (row >= nrows) return;
  const float4* w4 = (const float4*)(W + (size_t)row * HID);
  const float4* x4 = (const float4*)x;
  float acc = 0.f;
#pragma unroll 8
  for (int i = 0; i < 32; ++i) {
    float4 a = w4[i * 32 + lane];
    float4 b = x4[i * 32 + lane];
    acc += a.x * b.x + a.y * b.y + a.z * b.z + a.w * b.w;
  }
  for (int o = 16; o; o >>= 1) acc += __shfl_xor(acc, o, 32);
  if (lane == 0) y[row] = acc;
}

// ---------------- RoPE at pos = S, in-place on q and k ----------------
__global__ void rope_qk(float* __restrict__ q, float* __restrict__ k) {
  int h = blockIdx.x, d = threadIdx.x;                    // 32 blocks x 128
  int p = d & 63;
  bool hi = d >= 64;
  float inv = __expf(-(2.0f * (float)p / (float)DH) * logf(10000.0f));
  float ang = 32768.0f * inv;
  float s, c;
  __sincosf(ang, &s, &c);
  int base = h * DH;
  int pd = hi ? d - 64 : d + 64;
  float q0 = q[base + d], qp = q[base + pd];
  float k0 = k[base + d], kp = k[base + pd];
  __syncthreads();
  q[base + d] = hi ? (q0 * c + qp * s) : (q0 * c - qp * s);
  k[base + d] = hi ? (k0 * c + kp * s) : (k0 * c - kp * s);
}

// ---------------- centroid scores via V_WMMA_F32_16X16X4_F32 ----------------
// One wave computes 16 centroid dot-products (M=16 rows of A), B = q chunk
// broadcast across all N columns; D column 0 (lanes 0 / 16) holds the results.
__global__ void centroid_score(const float* __restrict__ cent,
                               const float* __restrict__ q,
                               float* __restrict__ cscore) {
  int wave = blockIdx.x * 4 + (threadIdx.x >> 5);   // 512 blocks * 4 waves = 2048
  int lane = threadIdx.x & 31;
  int h  = wave >> 6;                                // 64 groups of 16 per head
  int c0 = (wave & 63) << 4;
  int m  = lane & 15;
  int ko = (lane < 16) ? 0 : 2;
  const float* crow = cent + ((size_t)(h * NLIST + c0 + m)) * DH;
  const float* qh   = q + h * DH;
  v8f acc;
#pragma unroll
  for (int i = 0; i < 8; ++i) acc[i] = 0.f;
  for (int kb = 0; kb < DH; kb += 4) {
    v2f a, b;
    a.x = crow[kb + ko];     a.y = crow[kb + ko + 1];
    b.x = qh[kb + ko];       b.y = qh[kb + ko + 1];
    acc = __builtin_amdgcn_wmma_f32_16x16x4_f32(false, a, false, b,
                                                (short)0, acc, false, false);
  }
  if (lane == 0) {
#pragma unroll
    for (int r = 0; r < 8; ++r) cscore[h * NLIST + c0 + r] = acc[r];
  } else if (lane == 16) {
#pragma unroll
    for (int r = 0; r < 8; ++r) cscore[h * NLIST + c0 + 8 + r] = acc[r];
  }
}

// ---------------- exact top-NSEL per head: 3-level radix select ----------------
__device__ __forceinline__ unsigned score_key(const int* __restrict__ assign,
                                              const float* __restrict__ cscore,
                                              int h, int t) {
  int c = assign[(size_t)h * SMID + t];
  unsigned u = __float_as_uint(cscore[h * NLIST + c]);
  return (u & 0x80000000u) ? ~u : (u | 0x80000000u);   // monotonic key
}

__global__ void select_topk(const int* __restrict__ assign,
                            const float* __restrict__ cscore,
                            int* __restrict__ sel) {
  int h = blockIdx.x, tid = threadIdx.x;               // 32 blocks x 1024
  __shared__ unsigned hist[2048];
  __shared__ unsigned st[6];  // 0:b0 1:b1 2:b2 3:need 4:cntHi 5:cntEq

  // level 0: key[31:21]
  for (int i = tid; i < 2048; i += 1024) hist[i] = 0;
  __syncthreads();
  for (int t = tid; t < SMID; t += 1024)
    atomicAdd(&hist[score_key(assign, cscore, h, t) >> 21], 1u);
  __syncthreads();
  if (tid == 0) {
    unsigned cum = 0, need = NSEL;
    for (int b = 2047; b >= 0; --b) {
      if (cum + hist[b] >= need) { st[0] = b; st[3] = need - cum; break; }
      cum += hist[b];
    }
  }
  __syncthreads();
  unsigned b0 = st[0];
  // level 1: key[20:10] within bin b0
  for (int i = tid; i < 2048; i += 1024) hist[i] = 0;
  __syncthreads();
  for (int t = tid; t < SMID; t += 1024) {
    unsigned k = score_key(assign, cscore, h, t);
    if ((k >> 21) == b0) atomicAdd(&hist[(k >> 10) & 0x7FF], 1u);
  }
  __syncthreads();
  if (tid == 0) {
    unsigned cum = 0, need = st[3];
    for (int b = 2047; b >= 0; --b) {
      if (cum + hist[b] >= need) { st[1] = b; st[3] = need - cum; break; }
      cum += hist[b];
    }
  }
  __syncthreads();
  unsigned pre22 = (b0 << 11) | st[1];
  // level 2: key[9:0] within (b0,b1)
  for (int i = tid; i < 1024; i += 1024) hist[i] = 0;
  __syncthreads();
  for (int t = tid; t < SMID; t += 1024) {
    unsigned k = score_key(assign, cscore, h, t);
    if ((k >> 10) == pre22) atomicAdd(&hist[k & 0x3FF], 1u);
  }
  __syncthreads();
  if (tid == 0) {
    unsigned cum = 0, need = st[3];
    for (int b = 1023; b >= 0; --b) {
      if (cum + hist[b] >= need) { st[2] = b; st[3] = need - cum; break; }
      cum += hist[b];
    }
    st[4] = 0; st[5] = 0;
  }
  __syncthreads();
  unsigned Kt = (pre22 << 10) | st[2];
  unsigned need = st[3];
  // compaction: key > Kt from front, key == Kt (first `need`) from back
  for (int t = tid; t < SMID; t += 1024) {
    unsigned k = score_key(assign, cscore, h, t);
    if (k > Kt) {
      unsigned p = atomicAdd(&st[4], 1u);
      sel[h * NSEL + p] = t + SINK;
    } else if (k == Kt) {
      unsigned e = atomicAdd(&st[5], 1u);
      if (e < need) sel[h * NSEL + (NSEL - 1 - (int)e)] = t + SINK;
    }
  }
}

// ---------------- flash attention over selected tokens, split-K partials ----------------
__global__ void attn_partial(const float* __restrict__ kc, const float* __restrict__ vc,
                             const float* __restrict__ q,
                             const float* __restrict__ knew, const float* __restrict__ vnew,
                             const int* __restrict__ sel,
                             float* __restrict__ pm, float* __restrict__ pl,
                             float* __restrict__ pacc) {
  int h  = blockIdx.x >> 3;
  int sp = blockIdx.x & (NSPLIT - 1);
  int tid = threadIdx.x, w = tid >> 5, lane = tid & 31;
  const int chunk = (LTOT + NSPLIT - 1) / NSPLIT;      // 273
  int t0 = sp * chunk;
  int t1 = min(LTOT, t0 + chunk);

  float4 q4 = ((const float4*)(q + h * DH))[lane];
  float m = -3.0e38f, l = 0.f;
  float4 a = {0.f, 0.f, 0.f, 0.f};

  for (int t = t0 + w; t < t1; t += 8) {
    int row;
    if (t < SINK)              row = t;
    else if (t < SINK + NSEL)  row = sel[h * NSEL + (t - SINK)];
    else                       row = (SLEN - WIN) + (t - (SINK + NSEL));
    const float4 *Kr, *Vr;
    if (row == SLEN) { Kr = (const float4*)(knew + h * DH); Vr = (const float4*)(vnew + h * DH); }
    else {
      size_t base = ((size_t)row * NH + h) * DH;
      Kr = (const float4*)(kc + base); Vr = (const float4*)(vc + base);
    }
    float4 kv = Kr[lane];
    float s = q4.x * kv.x + q4.y * kv.y + q4.z * kv.z + q4.w * kv.w;
    for (int o = 16; o; o >>= 1) s += __shfl_xor(s, o, 32);
    s *= 0.08838834764831845f;                         // 1/sqrt(128)
    float mn = fmaxf(m, s);
    float sc = __expf(m - mn), p = __expf(s - mn);
    float4 vv = Vr[lane];
    a.x = a.x * sc + p * vv.x;  a.y = a.y * sc + p * vv.y;
    a.z = a.z * sc + p * vv.z;  a.w = a.w * sc + p * vv.w;
    l = l * sc + p;  m = mn;
  }

  __shared__ float sm[8], sl[8], sacc[8 * DH];
  if (lane == 0) { sm[w] = m; sl[w] = l; }
  ((float4*)&sacc[w * DH])[lane] = a;
  __syncthreads();
  if (tid < DH) {
    float M = -3.0e38f;
    for (int i = 0; i < 8; ++i) M = fmaxf(M, sm[i]);
    float acc = 0.f, L = 0.f;
    for (int i = 0; i < 8; ++i) {
      float e = __expf(sm[i] - M);
      acc += sacc[i * DH + tid] * e;
      L   += sl[i] * e;
    }
    pacc[(h * NSPLIT + sp) * DH + tid] = acc;
    if (tid == 0) { pm[h * NSPLIT + sp] = M; pl[h * NSPLIT + sp] = L; }
  }
}

__global__ void attn_combine(const float* __restrict__ pm, const float* __restrict__ pl,
                             const float* __restrict__ pacc, float* __restrict__ ctx) {
  int h = blockIdx.x, d = threadIdx.x;                 // 32 blocks x 128
  float M = -3.0e38f;
  for (int sp = 0; sp < NSPLIT; ++sp) M = fmaxf(M, pm[h * NSPLIT + sp]);
  float L = 0.f, acc = 0.f;
  for (int sp = 0; sp < NSPLIT; ++sp) {
    float e = __expf(pm[h * NSPLIT + sp] - M);
    L   += pl[h * NSPLIT + sp] * e;
    acc += pacc[(h * NSPLIT + sp) * DH + d] * e;
  }
  ctx[h * DH + d] = acc / L;
}

// ---------------- launcher ----------------
extern "C" void kernel_launch(void* const* d_in, const int* in_sizes, int n_in,
                              void* d_out, int out_size, void* d_ws, size_t ws_size,
                              hipStream_t stream) {
  const float* x    = (const float*)d_in[0];
  const float* Wq   = (const float*)d_in[1];
  const float* Wk   = (const float*)d_in[2];
  const float* Wv   = (const float*)d_in[3];
  const float* Wo   = (const float*)d_in[4];
  const float* kc   = (const float*)d_in[5];
  const float* vc   = (const float*)d_in[6];
  const float* cent = (const float*)d_in[7];
  const int*   asg  = (const int*)d_in[8];
  (void)in_sizes; (void)n_in; (void)out_size; (void)ws_size;

  float* ws   = (float*)d_ws;
  float* q    = ws;                     // 4096
  float* kn   = ws + 4096;              // 4096
  float* vn   = ws + 8192;              // 4096
  float* csc  = ws + 12288;             // 32*1024 = 32768
  int*   sel  = (int*)(ws + 45056);     // 32*2048 ints
  float* pm   = ws + 110592;            // 256
  float* pl   = ws + 110848;            // 256
  float* pacc = ws + 111104;            // 256*128 = 32768
  float* ctx  = ws + 143872;            // 4096

  gemv4096<<<512, 256, 0, stream>>>(Wq, x, q,  HID);
  gemv4096<<<512, 256, 0, stream>>>(Wk, x, kn, HID);
  gemv4096<<<512, 256, 0, stream>>>(Wv, x, vn, HID);
  rope_qk<<<NH, DH, 0, stream>>>(q, kn);
  centroid_score<<<512, 128, 0, stream>>>(cent, q, csc);
  select_topk<<<NH, 1024, 0, stream>>>(asg, csc, sel);
  attn_partial<<<NH * NSPLIT, 256, 0, stream>>>(kc, vc, q, kn, vn, sel, pm, pl, pacc);
  attn_combine<<<NH, DH, 0, stream>>>(pm, pl, pacc, ctx);
  gemv4096<<<512, 256, 0, stream>>>(Wo, ctx, (float*)d_out, HID);
}